// Renderer_9036611191496
// MI455X (gfx1250) — compile-verified
//
#include <hip/hip_runtime.h>
#include <math.h>

typedef __attribute__((ext_vector_type(2))) float v2f;
typedef __attribute__((ext_vector_type(8))) float v8f;
typedef __attribute__((ext_vector_type(4))) int   v4i;
typedef __attribute__((address_space(1))) v4i     g_v4i;   // global
typedef __attribute__((address_space(3))) v4i     l_v4i;   // LDS

#define SH_C0f 0.28209479177387814f
#define RECF   12          // floats per gaussian record: u v ia ib ic op r g b z pad pad
#define SORTN  4096        // pow2 LDS sort capacity (N=2048 fits)

#ifdef __HIP_DEVICE_COMPILE__
# if __has_builtin(__builtin_amdgcn_wmma_f32_16x16x4_f32)
#  define HAVE_WMMA 1
# endif
# if __has_builtin(__builtin_amdgcn_global_load_async_to_lds_b128)
#  define HAVE_ASYNC 1
# endif
#endif
#ifndef HAVE_WMMA
# define HAVE_WMMA 0
#endif
#ifndef HAVE_ASYNC
# define HAVE_ASYNC 0
#endif

// ---------------------------------------------------------------- preprocess
__global__ void gs_preprocess(const float* __restrict__ xyz,
                              const float* __restrict__ feat,
                              const float* __restrict__ scal,
                              const float* __restrict__ rotq,
                              const float* __restrict__ opac,
                              const float* __restrict__ c2w,
                              const float* __restrict__ intr,
                              const int*   __restrict__ pNear,
                              const int*   __restrict__ pFar,
                              float* __restrict__ recs,
                              int N, int N4)
{
  int i = blockIdx.x * blockDim.x + threadIdx.x;
  if (i >= N4) return;
  float* rec = recs + (size_t)i * RECF;
  if (i >= N) {                       // padding: never contributes, sorts last
    #pragma unroll
    for (int q = 0; q < RECF; ++q) rec[q] = 0.0f;
    rec[9] = 1e9f;
    return;
  }

  // ---- w2c = inv(c2w) (affine, last row [0,0,0,1]) via 3x3 adjugate
  float a00=c2w[0], a01=c2w[1], a02=c2w[2],  t0=c2w[3];
  float a10=c2w[4], a11=c2w[5], a12=c2w[6],  t1=c2w[7];
  float a20=c2w[8], a21=c2w[9], a22=c2w[10], t2=c2w[11];
  float det3 = a00*(a11*a22-a12*a21) - a01*(a10*a22-a12*a20) + a02*(a10*a21-a11*a20);
  float id = 1.0f / det3;
  float r00=(a11*a22-a12*a21)*id, r01=(a02*a21-a01*a22)*id, r02=(a01*a12-a02*a11)*id;
  float r10=(a12*a20-a10*a22)*id, r11=(a00*a22-a02*a20)*id, r12=(a02*a10-a00*a12)*id;
  float r20=(a10*a21-a11*a20)*id, r21=(a01*a20-a00*a21)*id, r22=(a00*a11-a01*a10)*id;
  float tw0 = -(r00*t0 + r01*t1 + r02*t2);
  float tw1 = -(r10*t0 + r11*t1 + r12*t2);
  float tw2 = -(r20*t0 + r21*t1 + r22*t2);

  // ---- quaternion -> rotation, M = R*diag(s), cov3d = M M^T
  float qw=rotq[i*4+0], qx=rotq[i*4+1], qy=rotq[i*4+2], qz=rotq[i*4+3];
  float qn = sqrtf(qw*qw+qx*qx+qy*qy+qz*qz) + 1e-8f;
  qw/=qn; qx/=qn; qy/=qn; qz/=qn;
  float R0=1.f-2.f*(qy*qy+qz*qz), R1=2.f*(qx*qy-qw*qz), R2=2.f*(qx*qz+qw*qy);
  float R3=2.f*(qx*qy+qw*qz), R4=1.f-2.f*(qx*qx+qz*qz), R5=2.f*(qy*qz-qw*qx);
  float R6=2.f*(qx*qz-qw*qy), R7=2.f*(qy*qz+qw*qx), R8=1.f-2.f*(qx*qx+qy*qy);
  float s0=scal[i*3+0], s1=scal[i*3+1], s2=scal[i*3+2];
  float m00=R0*s0, m01=R1*s1, m02=R2*s2;
  float m10=R3*s0, m11=R4*s1, m12=R5*s2;
  float m20=R6*s0, m21=R7*s1, m22=R8*s2;
  float c00=m00*m00+m01*m01+m02*m02;
  float c01=m00*m10+m01*m11+m02*m12;
  float c02=m00*m20+m01*m21+m02*m22;
  float c11=m10*m10+m11*m11+m12*m12;
  float c12=m10*m20+m11*m21+m12*m22;
  float c22=m20*m20+m21*m21+m22*m22;

  // ---- project
  float x0=xyz[i*3+0], x1=xyz[i*3+1], x2=xyz[i*3+2];
  float p0=r00*x0+r01*x1+r02*x2+tw0;
  float p1=r10*x0+r11*x1+r12*x2+tw1;
  float p2=r20*x0+r21*x1+r22*x2+tw2;
  float z = p2, zs = fmaxf(z, 1e-4f);
  float fx=intr[0], fy=intr[1], cx=intr[2], cy=intr[3];
  float u = fx*p0/zs + cx;
  float v = fy*p1/zs + cy;

  // ---- cov_cam = Rw C Rw^T
  float g00=r00*c00+r01*c01+r02*c02, g01=r00*c01+r01*c11+r02*c12, g02=r00*c02+r01*c12+r02*c22;
  float g10=r10*c00+r11*c01+r12*c02, g11=r10*c01+r11*c11+r12*c12, g12=r10*c02+r11*c12+r12*c22;
  float g20=r20*c00+r21*c01+r22*c02, g21=r20*c01+r21*c11+r22*c12, g22=r20*c02+r21*c12+r22*c22;
  float cc00=g00*r00+g01*r01+g02*r02;
  float cc01=g00*r10+g01*r11+g02*r12;
  float cc02=g00*r20+g01*r21+g02*r22;
  float cc11=g10*r10+g11*r11+g12*r12;
  float cc12=g10*r20+g11*r21+g12*r22;
  float cc22=g20*r20+g21*r21+g22*r22;

  // ---- cov2d = J cov_cam J^T ; J = [[fx/zs,0,-fx p0/zs^2],[0,fy/zs,-fy p1/zs^2]]
  float j00=fx/zs,  j02=-fx*p0/(zs*zs);
  float j11=fy/zs,  j12=-fy*p1/(zs*zs);
  float V00=cc00*j00+cc02*j02;
  float V01=cc01*j00+cc12*j02;
  float V02=cc02*j00+cc22*j02;
  float a2 = V00*j00 + V02*j02 + 0.3f;
  float b2 = V01*j11 + V02*j12;
  float c2v = (cc11*j11+cc12*j12)*j11 + (cc12*j11+cc22*j12)*j12 + 0.3f;
  float det2 = a2*c2v - b2*b2;

  float nearf = (float)pNear[0], farf = (float)pFar[0];
  bool valid = (z > nearf) && (z < farf) && (det2 > 1e-12f);
  float dd = (det2 > 1e-12f) ? det2 : 1.0f;
  float ia = c2v/dd, ib = -b2/dd, ic = a2/dd;

  float cr = fmaxf(SH_C0f*feat[i*3+0] + 0.5f, 0.f);
  float cg = fmaxf(SH_C0f*feat[i*3+1] + 0.5f, 0.f);
  float cb = fmaxf(SH_C0f*feat[i*3+2] + 0.5f, 0.f);
  float op = valid ? opac[i] : 0.0f;   // invalid -> alpha==0 everywhere

  rec[0]=u; rec[1]=v; rec[2]=ia; rec[3]=ib; rec[4]=ic; rec[5]=op;
  rec[6]=cr; rec[7]=cg; rec[8]=cb; rec[9]=z; rec[10]=0.f; rec[11]=0.f;
}

// ---------------------------------------------------------------- bitonic sort by z + gather
__global__ void gs_sort_gather(const float* __restrict__ recs,
                               float* __restrict__ srecs, int N4)
{
  __shared__ float key[SORTN];
  __shared__ int   idx[SORTN];
  int tid = threadIdx.x;                           // 1024 threads
  for (int i = tid; i < SORTN; i += 1024) {
    key[i] = (i < N4) ? recs[(size_t)i*RECF + 9] : 3.4e38f;
    idx[i] = i;
  }
  __syncthreads();
  for (int k = 2; k <= SORTN; k <<= 1) {
    for (int j = k >> 1; j > 0; j >>= 1) {
      for (int i = tid; i < SORTN; i += 1024) {
        int ixj = i ^ j;
        if (ixj > i) {
          bool up = ((i & k) == 0);
          float ki = key[i], kj = key[ixj];
          if ((ki > kj) == up) {
            key[i] = kj; key[ixj] = ki;
            int t = idx[i]; idx[i] = idx[ixj]; idx[ixj] = t;
          }
        }
      }
      __syncthreads();
    }
  }
  for (int s = tid; s < N4; s += 1024) {
    int src = idx[s];
    const float* rs = recs  + (size_t)src*RECF;
    float*       rd = srecs + (size_t)s  *RECF;
    #pragma unroll
    for (int q = 0; q < RECF; ++q) rd[q] = rs[q];
  }
}

// ---------------------------------------------------------------- render
__device__ __forceinline__ float gs_alpha(float u, float v, float ia, float ib,
                                          float ic, float op, float px, float py)
{
  float dx = px - u, dy = py - v;
  float pw = -0.5f*(ia*dx*dx + ic*dy*dy) - ib*dx*dy;
  float al = fminf(op * __expf(fminf(pw, 0.0f)), 0.99f);
  return (pw <= 0.0f && al >= (1.0f/255.0f)) ? al : 0.0f;
}

__global__ __launch_bounds__(256)
void gs_render(const float* __restrict__ srecs,
               const int*   __restrict__ pW,
               float* __restrict__ out, int N4, int pixels)
{
  extern __shared__ float lds_rec[];           // N4*RECF floats (96KB @ N=2048, <320KB/WGP)

  // ---- stage sorted records into LDS once per workgroup (async global->LDS)
  {
    const int nChunks = (N4 * RECF) >> 2;      // 16-byte chunks
    const char* gsrc = (const char*)srecs;
    char*       ldst = (char*)lds_rec;
    for (int ch = threadIdx.x; ch < nChunks; ch += blockDim.x) {
#if HAVE_ASYNC
      __builtin_amdgcn_global_load_async_to_lds_b128(
          (g_v4i*)(gsrc + (size_t)ch * 16),
          (l_v4i*)(ldst + (size_t)ch * 16),
          0, 0);
#else
      ((float4*)ldst)[ch] = ((const float4*)gsrc)[ch];
#endif
    }
#if HAVE_ASYNC
# if __has_builtin(__builtin_amdgcn_s_wait_asynccnt)
    __builtin_amdgcn_s_wait_asynccnt(0);
# else
    asm volatile("s_wait_asynccnt 0" ::: "memory");
# endif
#endif
    __syncthreads();
  }

  const int W    = pW[0];
  const int lane = threadIdx.x & 31;
  const int wv   = threadIdx.x >> 5;
  const int gw   = blockIdx.x * (blockDim.x >> 5) + wv;
  const int pixBase = gw * 32;                 // 32 pixels per wave (two 16-wide tiles)
  if (pixBase >= pixels) return;               // wave-uniform: EXEC stays all-ones
  const int py  = pixBase / W;
  const int px0 = pixBase - py * W;
  const int m    = lane & 15;                  // pixel within tile / channel column
  const int half = lane >> 4;                  // 0: gaussians k,k+1 ; 1: k+2,k+3
  const float pcy  = (float)py + 0.5f;
  const float pcx0 = (float)(px0 + m) + 0.5f;
  const float pcx1 = pcx0 + 16.0f;

  float T0 = 1.0f, T1 = 1.0f;                  // transmittance, per pixel (dup per half)
#if HAVE_WMMA
  v8f acc0 = {}; v8f acc1 = {};
#else
  float f0[4] = {0,0,0,0}, f1[4] = {0,0,0,0};
#endif

  for (int base = 0; base < N4; base += 4) {
    const float* r0 = lds_rec + (size_t)(base + half*2) * RECF;  // ds_load path
    const float* r1 = r0 + RECF;
    float u0=r0[0], v0=r0[1], ia0=r0[2], ib0=r0[3], ic0=r0[4], op0=r0[5];
    float cr0=r0[6], cg0=r0[7], cb0=r0[8], z0=r0[9];
    float u1=r1[0], v1=r1[1], ia1=r1[2], ib1=r1[3], ic1=r1[4], op1=r1[5];
    float cr1=r1[6], cg1=r1[7], cb1=r1[8], z1=r1[9];

    float a00 = gs_alpha(u0,v0,ia0,ib0,ic0,op0, pcx0, pcy);
    float a01 = gs_alpha(u1,v1,ia1,ib1,ic1,op1, pcx0, pcy);
    float a10 = gs_alpha(u0,v0,ia0,ib0,ic0,op0, pcx1, pcy);
    float a11 = gs_alpha(u1,v1,ia1,ib1,ic1,op1, pcx1, pcy);

    float pa0 = (1.f - a00) * (1.f - a01);     // survive-product for my 2 gaussians
    float pa1 = (1.f - a10) * (1.f - a11);
    float paO0 = __shfl_xor(pa0, 16, 32);      // other half's product
    float paO1 = __shfl_xor(pa1, 16, 32);
    float Ts0 = half ? T0 * paO0 : T0;         // carry-in for upper half
    float Ts1 = half ? T1 * paO1 : T1;
    float w00 = Ts0 * a00, w01 = Ts0 * (1.f - a00) * a01;
    float w10 = Ts1 * a10, w11 = Ts1 * (1.f - a10) * a11;
    T0 *= pa0 * paO0;
    T1 *= pa1 * paO1;

    // B fragment: 4x16 (gaussian x channel), cols 0..3 = r,g,b,z
    float bx = (m==0)?cr0 : (m==1)?cg0 : (m==2)?cb0 : (m==3)?z0 : 0.f;
    float by = (m==0)?cr1 : (m==1)?cg1 : (m==2)?cb1 : (m==3)?z1 : 0.f;

#if HAVE_WMMA
    v2f af0; af0.x = w00; af0.y = w01;         // A 16x4 f32 lane layout
    v2f af1; af1.x = w10; af1.y = w11;
    v2f bf;  bf.x  = bx;  bf.y  = by;
    acc0 = __builtin_amdgcn_wmma_f32_16x16x4_f32(false, af0, false, bf,
                                                 (short)0, acc0, false, false);
    acc1 = __builtin_amdgcn_wmma_f32_16x16x4_f32(false, af1, false, bf,
                                                 (short)0, acc1, false, false);
#else
    #pragma unroll
    for (int c = 0; c < 4; ++c) {
      int src = (half << 4) | c;
      float B0 = __shfl(bx, src, 32);
      float B1 = __shfl(by, src, 32);
      f0[c] += w00 * B0 + w01 * B1;
      f1[c] += w10 * B0 + w11 * B1;
    }
#endif
  }

#if HAVE_WMMA
  // D layout: VGPR j holds M=j (lanes 0-15) / M=j+8 (lanes 16-31), N=lane&15
  if (m < 4) {
    size_t plane = (size_t)m * pixels + (size_t)py * W + px0;
    #pragma unroll
    for (int j = 0; j < 8; ++j) {
      int M = half * 8 + j;
      out[plane + M]      = acc0[j];
      out[plane + 16 + M] = acc1[j];
    }
  }
#else
  #pragma unroll
  for (int c = 0; c < 4; ++c) {
    f0[c] += __shfl_xor(f0[c], 16, 32);
    f1[c] += __shfl_xor(f1[c], 16, 32);
  }
  if (half == 0) {
    #pragma unroll
    for (int c = 0; c < 4; ++c) {
      size_t plane = (size_t)c * pixels + (size_t)py * W + px0;
      out[plane + m]      = f0[c];
      out[plane + 16 + m] = f1[c];
    }
  }
#endif
}

// ---------------------------------------------------------------- host
extern "C" void kernel_launch(void* const* d_in, const int* in_sizes, int n_in,
                              void* d_out, int out_size, void* d_ws, size_t ws_size,
                              hipStream_t stream)
{
  const float* xyz  = (const float*)d_in[0];
  const float* feat = (const float*)d_in[1];
  const float* scal = (const float*)d_in[2];
  const float* rot  = (const float*)d_in[3];
  const float* opac = (const float*)d_in[4];
  const float* c2w  = (const float*)d_in[5];
  const float* intr = (const float*)d_in[6];
  const int*   pW   = (const int*)d_in[7];
  const int*   pNear= (const int*)d_in[10];
  const int*   pFar = (const int*)d_in[11];

  int N  = in_sizes[0] / 3;                    // (B=1, N, 3)
  int N4 = (N + 3) & ~3;

  float* recs  = (float*)d_ws;                 // N4 * 12 floats
  float* srecs = recs + (size_t)N4 * RECF;     // N4 * 12 floats (sorted)

  gs_preprocess<<<(N4 + 255) / 256, 256, 0, stream>>>(
      xyz, feat, scal, rot, opac, c2w, intr, pNear, pFar, recs, N, N4);

  gs_sort_gather<<<1, 1024, 0, stream>>>(recs, srecs, N4);

  int pixels = out_size / 4;                   // (1,1,4,H,W)
  int waves  = (pixels + 31) / 32;
  int blocks = (waves + 7) / 8;                // 8 waves (256 threads) per block
  size_t shbytes = (size_t)N4 * RECF * sizeof(float);   // 96KB @ N=2048
  gs_render<<<blocks, 256, shbytes, stream>>>(srecs, pW, (float*)d_out, N4, pixels);
}